// _ResidualBlock_59906203845002
// MI455X (gfx1250) — compile-verified
//
#include <hip/hip_runtime.h>
#include <math.h>

typedef __attribute__((ext_vector_type(2))) float v2f;
typedef __attribute__((ext_vector_type(8))) float v8f;

#define TIN 256   // IN
#define TOUT 128  // OUT
#define NHEAD 4
#define NCH 32
#define NGRAPH 64
#define GN_EPS 1e-5f

// ---------------------------------------------------------------------------
// K1: fused triple GEMM (xl = x@Wl, xr = x@Wr, xs = x@Ws) using f32 WMMA.
// Block = 256 threads (8 waves), one block per 16-row tile of x. The A-tile
// (16x256 f32) is staged in LDS ONCE and shared by all 8 waves AND all three
// weight matrices: each wave owns a 16-col slice and keeps 3 accumulators,
// issuing 3 V_WMMA_F32_16X16X4_F32 per K-step (192 WMMAs per wave total).
// This guarantees a single HBM pass over x (102 MB) for all three GEMMs.
// ---------------------------------------------------------------------------
__global__ __launch_bounds__(256) void gemm3_wmma_f32(
    const float* __restrict__ x,
    const float* __restrict__ Wl,
    const float* __restrict__ Wr,
    const float* __restrict__ Ws,
    float* __restrict__ xl,
    float* __restrict__ xr,
    float* __restrict__ xs,
    int N)
{
    __shared__ float As[16][TIN + 4];   // +4 floats pad: conflict-free, keeps 16B align

    const int rowBase = blockIdx.x * 16;

    // Cooperative A-tile load: 16 rows x 256 f32 = 4096 floats, 16 per thread.
    {
        const int t  = threadIdx.x;
        const int r  = t >> 4;
        const int c0 = (t & 15) * 16;
        const float4* __restrict__ src = (const float4*)(x + (size_t)(rowBase + r) * TIN + c0);
        float4* dst4 = (float4*)(&As[r][c0]);
#pragma unroll
        for (int j = 0; j < 4; ++j) dst4[j] = src[j];
    }
    __syncthreads();

    const int lane    = threadIdx.x & 31;
    const int wave    = threadIdx.x >> 5;
    const int colBase = wave * 16;             // 8 waves * 16 = 128 cols
    const int nIdx    = colBase + (lane & 15);
    const int arow    = lane & 15;
    const int koff    = (lane < 16) ? 0 : 2;   // K interleave: halves of the wave

    v8f accL = {}, accR = {}, accS = {};
#pragma unroll 4
    for (int kk = 0; kk < TIN; kk += 4) {
        // A fragment (16x4): VGPR0 = K+koff, VGPR1 = K+koff+1 (per ISA layout)
        float2 a2 = *(const float2*)&As[arow][kk + koff];
        v2f A; A.x = a2.x; A.y = a2.y;
        const size_t o0 = (size_t)(kk + koff) * TOUT + nIdx;
        // B fragments (4x16) for all three weight matrices, same K interleave
        v2f BL, BR, BS;
        BL.x = Wl[o0]; BL.y = Wl[o0 + TOUT];
        BR.x = Wr[o0]; BR.y = Wr[o0 + TOUT];
        BS.x = Ws[o0]; BS.y = Ws[o0 + TOUT];
        accL = __builtin_amdgcn_wmma_f32_16x16x4_f32(false, A, false, BL, (short)0, accL, false, false);
        accR = __builtin_amdgcn_wmma_f32_16x16x4_f32(false, A, false, BR, (short)0, accR, false, false);
        accS = __builtin_amdgcn_wmma_f32_16x16x4_f32(false, A, false, BS, (short)0, accS, false, false);
    }

    // C/D layout: VGPR v -> M = v + (lane<16 ? 0 : 8), N = lane&15
    const int mBase = (lane < 16) ? 0 : 8;
    const int col   = colBase + (lane & 15);
#pragma unroll
    for (int v = 0; v < 8; ++v) {
        const size_t o = (size_t)(rowBase + mBase + v) * TOUT + col;
        xl[o] = accL[v];
        xr[o] = accR[v];
        xs[o] = accS[v];
    }
}

// ---------------------------------------------------------------------------
// Order-preserving float <-> uint encoding for atomic max on f32
// ---------------------------------------------------------------------------
__device__ __forceinline__ unsigned fenc(float f) {
    int b = __float_as_int(f);
    return (unsigned)b ^ (unsigned)((b >> 31) | 0x80000000);
}
__device__ __forceinline__ float fdec(unsigned u) {
    int b = (u & 0x80000000u) ? (int)(u ^ 0x80000000u) : ~(int)u;
    return __int_as_float(b);
}

// ---------------------------------------------------------------------------
// K2: per (edge, head): logit = att[h] . leaky_relu(xl[src,h,:] + xr[dst,h,:])
//     + segment max via encoded atomicMax. Self-loops appended at e >= E.
// ---------------------------------------------------------------------------
__global__ void edge_logits(
    const float* __restrict__ xl, const float* __restrict__ xr,
    const float* __restrict__ att, const int* __restrict__ ei,
    float* __restrict__ logits, unsigned* __restrict__ lmax,
    int E, int EN)
{
    int idx = blockIdx.x * blockDim.x + threadIdx.x;
    if (idx >= EN * NHEAD) return;
    const int h = idx & (NHEAD - 1);
    const int e = idx >> 2;
    int src, dst;
    if (e < E) { src = ei[e]; dst = ei[E + e]; }
    else       { src = dst = e - E; }

    const float4* __restrict__ pl = (const float4*)(xl + (size_t)src * TOUT + h * NCH);
    const float4* __restrict__ pr = (const float4*)(xr + (size_t)dst * TOUT + h * NCH);
    const float4* __restrict__ pa = (const float4*)(att + h * NCH);

    float s = 0.f;
#pragma unroll
    for (int j = 0; j < NCH / 4; ++j) {
        float4 a = pl[j], b = pr[j], w = pa[j];
        float vx = a.x + b.x; vx = vx > 0.f ? vx : 0.2f * vx;
        float vy = a.y + b.y; vy = vy > 0.f ? vy : 0.2f * vy;
        float vz = a.z + b.z; vz = vz > 0.f ? vz : 0.2f * vz;
        float vw = a.w + b.w; vw = vw > 0.f ? vw : 0.2f * vw;
        s += vx * w.x + vy * w.y + vz * w.z + vw * w.w;
    }
    logits[idx] = s;
    atomicMax(&lmax[(size_t)dst * NHEAD + h], fenc(s));
}

// ---------------------------------------------------------------------------
// K3: ex = exp(logit - lmax[dst]); denom[dst] += ex  (logits overwritten)
// ---------------------------------------------------------------------------
__global__ void edge_softmax(
    const int* __restrict__ ei,
    float* __restrict__ logits,
    const unsigned* __restrict__ lmax,
    float* __restrict__ denom,
    int E, int EN)
{
    int idx = blockIdx.x * blockDim.x + threadIdx.x;
    if (idx >= EN * NHEAD) return;
    const int h = idx & (NHEAD - 1);
    const int e = idx >> 2;
    int dst = (e < E) ? ei[E + e] : (e - E);
    float ex = __expf(logits[idx] - fdec(lmax[(size_t)dst * NHEAD + h]));
    logits[idx] = ex;
    atomicAdd(&denom[(size_t)dst * NHEAD + h], ex);
}

// ---------------------------------------------------------------------------
// K4: out[dst] += (ex/denom[dst]) * xl[src]; thread = (edge, head, quarter)
// ---------------------------------------------------------------------------
__global__ void edge_scatter(
    const float* __restrict__ xl, const int* __restrict__ ei,
    const float* __restrict__ ex, const float* __restrict__ denom,
    float* __restrict__ gat, int E, int EN)
{
    int idx = blockIdx.x * blockDim.x + threadIdx.x;
    if (idx >= EN * NHEAD * 4) return;
    const int q = idx & 3;
    const int h = (idx >> 2) & (NHEAD - 1);
    const int e = idx >> 4;
    int src, dst;
    if (e < E) { src = ei[e]; dst = ei[E + e]; }
    else       { src = dst = e - E; }

    const float alpha = ex[(size_t)e * NHEAD + h] / denom[(size_t)dst * NHEAD + h];
    const float4* __restrict__ ps = (const float4*)(xl + (size_t)src * TOUT + h * NCH + q * 8);
    float4 v0 = ps[0], v1 = ps[1];
    float* pd = gat + (size_t)dst * TOUT + h * NCH + q * 8;
    atomicAdd(pd + 0, alpha * v0.x);
    atomicAdd(pd + 1, alpha * v0.y);
    atomicAdd(pd + 2, alpha * v0.z);
    atomicAdd(pd + 3, alpha * v0.w);
    atomicAdd(pd + 4, alpha * v1.x);
    atomicAdd(pd + 5, alpha * v1.y);
    atomicAdd(pd + 6, alpha * v1.z);
    atomicAdd(pd + 7, alpha * v1.w);
}

// ---------------------------------------------------------------------------
// K5: node counts per graph
// ---------------------------------------------------------------------------
__global__ void node_counts(const int* __restrict__ batch, float* __restrict__ counts, int N)
{
    int n = blockIdx.x * blockDim.x + threadIdx.x;
    if (n < N) atomicAdd(&counts[batch[n]], 1.0f);
}

// ---------------------------------------------------------------------------
// K6: add biases in-place + per-graph channel sums (run-length-flushed atomics;
//     batch is sorted, so consecutive nodes share a graph).
// ---------------------------------------------------------------------------
__global__ __launch_bounds__(TOUT) void bias_and_sums(
    float* __restrict__ gat, float* __restrict__ xs,
    const float* __restrict__ cbias, const float* __restrict__ sbias,
    const int* __restrict__ batch,
    float* __restrict__ sums1, float* __restrict__ sums2, int N)
{
    const int c  = threadIdx.x;
    const int n0 = blockIdx.x * 128;
    const float b1 = cbias[c], b2 = sbias[c];
    float s1 = 0.f, s2 = 0.f;
    int curg = -1;
    for (int i = 0; i < 128; ++i) {
        int n = n0 + i;
        if (n >= N) break;
        int g = batch[n];
        if (g != curg) {
            if (curg >= 0) {
                atomicAdd(&sums1[curg * TOUT + c], s1);
                atomicAdd(&sums2[curg * TOUT + c], s2);
            }
            curg = g; s1 = 0.f; s2 = 0.f;
        }
        size_t o = (size_t)n * TOUT + c;
        float h1 = gat[o] + b1;
        float h2 = xs[o] + b2;
        gat[o] = h1; xs[o] = h2;
        s1 += h1; s2 += h2;
    }
    if (curg >= 0) {
        atomicAdd(&sums1[curg * TOUT + c], s1);
        atomicAdd(&sums2[curg * TOUT + c], s2);
    }
}

// ---------------------------------------------------------------------------
// K7: per-graph variance of (h - mean_scale * mean), same run-flush structure
// ---------------------------------------------------------------------------
__global__ __launch_bounds__(TOUT) void var_sums(
    const float* __restrict__ gat, const float* __restrict__ xs,
    const float* __restrict__ ms1, const float* __restrict__ ms2,
    const int* __restrict__ batch, const float* __restrict__ counts,
    const float* __restrict__ sums1, const float* __restrict__ sums2,
    float* __restrict__ var1, float* __restrict__ var2, int N)
{
    const int c  = threadIdx.x;
    const int n0 = blockIdx.x * 128;
    const float m1s = ms1[c], m2s = ms2[c];
    float v1 = 0.f, v2 = 0.f;
    int curg = -1;
    float mean1 = 0.f, mean2 = 0.f;
    for (int i = 0; i < 128; ++i) {
        int n = n0 + i;
        if (n >= N) break;
        int g = batch[n];
        if (g != curg) {
            if (curg >= 0) {
                atomicAdd(&var1[curg * TOUT + c], v1);
                atomicAdd(&var2[curg * TOUT + c], v2);
            }
            curg = g; v1 = 0.f; v2 = 0.f;
            float cnt = fmaxf(counts[g], 1.0f);
            mean1 = sums1[g * TOUT + c] / cnt;
            mean2 = sums2[g * TOUT + c] / cnt;
        }
        size_t o = (size_t)n * TOUT + c;
        float d1 = gat[o] - m1s * mean1;
        float d2 = xs[o]  - m2s * mean2;
        v1 += d1 * d1; v2 += d2 * d2;
    }
    if (curg >= 0) {
        atomicAdd(&var1[curg * TOUT + c], v1);
        atomicAdd(&var2[curg * TOUT + c], v2);
    }
}

// ---------------------------------------------------------------------------
// K8: final normalize both branches, add, ELU
// ---------------------------------------------------------------------------
__global__ void finalize(
    const float* __restrict__ gat, const float* __restrict__ xs,
    const int* __restrict__ batch, const float* __restrict__ counts,
    const float* __restrict__ sums1, const float* __restrict__ sums2,
    const float* __restrict__ var1, const float* __restrict__ var2,
    const float* __restrict__ w1, const float* __restrict__ b1, const float* __restrict__ ms1,
    const float* __restrict__ w2, const float* __restrict__ b2, const float* __restrict__ ms2,
    float* __restrict__ out, int N)
{
    int idx = blockIdx.x * blockDim.x + threadIdx.x;
    if (idx >= N * TOUT) return;
    const int c = idx & (TOUT - 1);
    const int n = idx >> 7;
    const int g = batch[n];
    const float cnt = fmaxf(counts[g], 1.0f);

    float mean1 = sums1[g * TOUT + c] / cnt;
    float mean2 = sums2[g * TOUT + c] / cnt;
    float d1 = gat[idx] - ms1[c] * mean1;
    float d2 = xs[idx]  - ms2[c] * mean2;
    float iv1 = rsqrtf(var1[g * TOUT + c] / cnt + GN_EPS);
    float iv2 = rsqrtf(var2[g * TOUT + c] / cnt + GN_EPS);
    float y = (w1[c] * d1 * iv1 + b1[c]) + (w2[c] * d2 * iv2 + b2[c]);
    out[idx] = y > 0.f ? y : (__expf(y) - 1.f);
}

// ---------------------------------------------------------------------------
// Host launcher
// ---------------------------------------------------------------------------
extern "C" void kernel_launch(void* const* d_in, const int* in_sizes, int n_in,
                              void* d_out, int out_size, void* d_ws, size_t ws_size,
                              hipStream_t stream)
{
    const float* x      = (const float*)d_in[0];
    const float* W_l    = (const float*)d_in[1];
    const float* W_r    = (const float*)d_in[2];
    const float* att    = (const float*)d_in[3];
    const float* cbias  = (const float*)d_in[4];
    const float* skip_W = (const float*)d_in[5];
    const float* sbias  = (const float*)d_in[6];
    const float* gn1_w  = (const float*)d_in[7];
    const float* gn1_b  = (const float*)d_in[8];
    const float* gn1_ms = (const float*)d_in[9];
    const float* gn2_w  = (const float*)d_in[10];
    const float* gn2_b  = (const float*)d_in[11];
    const float* gn2_ms = (const float*)d_in[12];
    const int*   ei     = (const int*)d_in[13];
    const int*   batch  = (const int*)d_in[14];
    float*       out    = (float*)d_out;

    const int N  = in_sizes[14];
    const int E  = in_sizes[13] / 2;
    const int EN = E + N;

    // Workspace layout
    float* ws      = (float*)d_ws;
    float* xl      = ws;
    float* xr      = xl + (size_t)N * TOUT;
    float* xs      = xr + (size_t)N * TOUT;
    float* logits  = xs + (size_t)N * TOUT;
    float* gat     = logits + (size_t)EN * NHEAD;          // zeroed region starts here
    unsigned* lmax = (unsigned*)(gat + (size_t)N * TOUT);
    float* denom   = (float*)(lmax + (size_t)N * NHEAD);
    float* counts  = denom + (size_t)N * NHEAD;
    float* sums1   = counts + NGRAPH;
    float* sums2   = sums1 + NGRAPH * TOUT;
    float* var1    = sums2 + NGRAPH * TOUT;
    float* var2    = var1 + NGRAPH * TOUT;
    float* ws_end  = var2 + NGRAPH * TOUT;

    // Zero all accumulators (gat, lmax(=ordered -inf sentinel), denom, counts, stats)
    hipMemsetAsync(gat, 0, (char*)ws_end - (char*)gat, stream);

    // K1: fused triple GEMM (WMMA f32), single HBM pass over x
    gemm3_wmma_f32<<<(N + 15) / 16, 256, 0, stream>>>(x, W_l, W_r, skip_W, xl, xr, xs, N);

    // K5: counts (independent of GEMM, stream-serialized anyway)
    node_counts<<<(N + 255) / 256, 256, 0, stream>>>(batch, counts, N);

    // K2: logits + segment max
    {
        int threads = EN * NHEAD;
        edge_logits<<<(threads + 255) / 256, 256, 0, stream>>>(xl, xr, att, ei, logits, lmax, E, EN);
    }
    // K3: exp + denom
    {
        int threads = EN * NHEAD;
        edge_softmax<<<(threads + 255) / 256, 256, 0, stream>>>(ei, logits, lmax, denom, E, EN);
    }
    // K4: weighted scatter
    {
        int threads = EN * NHEAD * 4;
        edge_scatter<<<(threads + 255) / 256, 256, 0, stream>>>(xl, ei, logits, denom, gat, E, EN);
    }
    // K6: biases + per-graph sums
    bias_and_sums<<<(N + 127) / 128, TOUT, 0, stream>>>(gat, xs, cbias, sbias, batch, sums1, sums2, N);
    // K7: per-graph variances
    var_sums<<<(N + 127) / 128, TOUT, 0, stream>>>(gat, xs, gn1_ms, gn2_ms, batch, counts,
                                                   sums1, sums2, var1, var2, N);
    // K8: finalize
    {
        int threads = N * TOUT;
        finalize<<<(threads + 255) / 256, 256, 0, stream>>>(gat, xs, batch, counts,
                                                            sums1, sums2, var1, var2,
                                                            gn1_w, gn1_b, gn1_ms,
                                                            gn2_w, gn2_b, gn2_ms,
                                                            out, N);
    }
}